// LinearAttention_13056700580412
// MI455X (gfx1250) — compile-verified
//
#include <hip/hip_runtime.h>
#include <math.h>
#include <stdint.h>

typedef __attribute__((ext_vector_type(2))) float v2f;
typedef __attribute__((ext_vector_type(8))) float v8f;

#define B_ 8
#define C_ 256
#define KC_ 32
#define N_ 16384
#define NCHUNK 32                 // chunks per batch
#define CHUNKPX (N_ / NCHUNK)     // 512 pixels per chunk
#define TPX 32                    // pixels per LDS tile
#define TITER (CHUNKPX / TPX)     // 16 tile iterations per chunk

#define XS_LD 36
#define VS_LD 36
#define KS_LD 36

__device__ __forceinline__ v8f wmma4(v2f a, v2f b, v8f c) {
  // D = A(16x4,f32) * B(4x16,f32) + C(16x16,f32)
  return __builtin_amdgcn_wmma_f32_16x16x4_f32(false, a, false, b, (short)0, c,
                                               false, false);
}

__device__ __forceinline__ float fm_elu1(float t) {
  return (t > 0.0f) ? (t + 1.0f) : __expf(t);
}

// Async DMA: 16 bytes global -> LDS per lane (gfx1250 GLOBAL_LOAD_ASYNC_TO_LDS,
// GVS addressing: saddr base + 32-bit per-lane voffset). LDS byte offset is the
// low 32 bits of the generic pointer.
__device__ __forceinline__ void async_load16(const float* gbase, int goff_bytes,
                                             float* lds_ptr) {
  unsigned lds_off = (unsigned)(uintptr_t)lds_ptr;
  asm volatile("global_load_async_to_lds_b128 %0, %1, %2"
               :
               : "v"(lds_off), "v"(goff_bytes), "s"(gbase)
               : "memory");
}

__device__ __forceinline__ void wait_async0() {
  asm volatile("s_wait_asynccnt 0x0" ::: "memory");
}

// Stage x tile [256][TPX] into LDS via async DMA (8 x b128 per thread).
__device__ __forceinline__ void stage_x_tile(const float* xb, int base,
                                             float* xs, int tid) {
  for (int f = tid; f < C_ * (TPX / 4); f += 256) {
    const int row = f >> 3;
    const int c4 = (f & 7) * 4;
    async_load16(xb, (int)(((long)row * N_ + base + c4) * sizeof(float)),
                 &xs[row * XS_LD + c4]);
  }
  wait_async0();
}

// ---------------------------------------------------------------------------
// Kernel 1: per (batch, chunk) compute partial kv[32][256] and z[32]
// ---------------------------------------------------------------------------
__global__ __launch_bounds__(256) void k1_kv(
    const float* __restrict__ x, const float* __restrict__ Wk,
    const float* __restrict__ bk, const float* __restrict__ Wv,
    const float* __restrict__ bv, float* __restrict__ pkv,
    float* __restrict__ pz) {
  __shared__ float xs[C_ * XS_LD];    // x tile [256][32]
  __shared__ float vs[128 * VS_LD];   // v half-tile [128][32]
  __shared__ float ks[KC_ * KS_LD];   // km tile [32][32]
  __shared__ float zacc[KC_];

  const int tid = threadIdx.x;
  const int lane = tid & 31;
  const int wave = tid >> 5;
  const int chunk = blockIdx.x;
  const int b = blockIdx.y;
  const int n0 = chunk * CHUNKPX;

  if (tid < KC_) zacc[tid] = 0.0f;

  const int am = lane & 15;        // A-fragment M
  const int kg = (lane >> 4) * 2;  // K sub-offset for both A and B fragments
  const int nn = lane & 15;        // B/C/D fragment N
  const int mh = (lane >> 4) * 8;  // C/D fragment M-half offset

  v8f kvacc[4] = {};               // wave-persistent kv accumulators
  const int rt = wave & 1;         // kc row tile (0..1)
  const int c0 = wave >> 1;        // base vrow col-tile (0..3); owns c0+4j

  const float* xb = x + (long)b * C_ * N_;

  for (int t = 0; t < TITER; ++t) {
    __syncthreads();
    stage_x_tile(xb, n0 + t * TPX, xs, tid);
    __syncthreads();
    // ---- km = fm(Wk @ xs + bk): 4 tiles, waves 0-3 ----
    if (wave < 4) {
      const int mb = 16 * (wave & 1);
      const int nb = 16 * (wave >> 1);
      v8f acc = {};
      for (int kb = 0; kb < C_; kb += 4) {
        const int k = kb + kg;
        v2f a, bf;
        a.x = Wk[(mb + am) * C_ + k];
        a.y = Wk[(mb + am) * C_ + k + 1];
        bf.x = xs[k * XS_LD + nb + nn];
        bf.y = xs[(k + 1) * XS_LD + nb + nn];
        acc = wmma4(a, bf, acc);
      }
      for (int r = 0; r < 8; ++r) {
        const int m = mb + mh + r;
        ks[m * KS_LD + nb + nn] = fm_elu1(acc[r] + bk[m]);
      }
    }
    __syncthreads();
    // ---- z accumulation (fixed order) ----
    if (tid < KC_) {
      float s = 0.0f;
      for (int n = 0; n < TPX; ++n) s += ks[tid * KS_LD + n];
      zacc[tid] += s;
    }
    // ---- v halves: v = Wv @ xs + bv, then kv += km @ v^T ----
    for (int h = 0; h < 2; ++h) {
      for (int j = 0; j < 2; ++j) {  // 16 tiles / 8 waves
        const int tt = wave * 2 + j;
        const int mbG = 128 * h + 16 * (tt >> 1);
        const int nb = 16 * (tt & 1);
        v8f acc = {};
        for (int kb = 0; kb < C_; kb += 4) {
          const int k = kb + kg;
          v2f a, bf;
          a.x = Wv[(mbG + am) * C_ + k];
          a.y = Wv[(mbG + am) * C_ + k + 1];
          bf.x = xs[k * XS_LD + nb + nn];
          bf.y = xs[(k + 1) * XS_LD + nb + nn];
          acc = wmma4(a, bf, acc);
        }
        for (int r = 0; r < 8; ++r) {
          const int m = mbG + mh + r;
          vs[(m - 128 * h) * VS_LD + nb + nn] = acc[r] + bv[m];
        }
      }
      __syncthreads();
      // kv tiles owned by this wave inside half h: j2 in {2h, 2h+1}
      for (int j2 = 2 * h; j2 < 2 * h + 2; ++j2) {
        const int ct = c0 + 4 * j2;
        const int vb = 16 * ct - 128 * h;  // local vs row base
        v8f acc = kvacc[j2];
        for (int kb = 0; kb < TPX; kb += 4) {
          const int k = kb + kg;
          v2f a, bf;
          a.x = ks[(16 * rt + am) * KS_LD + k];
          a.y = ks[(16 * rt + am) * KS_LD + k + 1];
          bf.x = vs[(vb + nn) * VS_LD + k];
          bf.y = vs[(vb + nn) * VS_LD + k + 1];
          acc = wmma4(a, bf, acc);
        }
        kvacc[j2] = acc;
      }
      if (h == 0) __syncthreads();  // protect vs before half-1 overwrite
    }
  }
  __syncthreads();
  // ---- write partials ----
  float* pk = pkv + (long)(b * NCHUNK + chunk) * (KC_ * C_);
  for (int j = 0; j < 4; ++j) {
    const int ct = c0 + 4 * j;
    for (int r = 0; r < 8; ++r) {
      const int m = 16 * rt + mh + r;
      pk[m * C_ + 16 * ct + nn] = kvacc[j][r];
    }
  }
  if (tid < KC_) pz[(b * NCHUNK + chunk) * KC_ + tid] = zacc[tid];
}

// ---------------------------------------------------------------------------
// Kernel 2: deterministic fixed-order reduction of chunk partials
// ---------------------------------------------------------------------------
__global__ void k2_reduce(const float* __restrict__ pkv,
                          const float* __restrict__ pz, float* __restrict__ kvf,
                          float* __restrict__ zf) {
  const int idx = blockIdx.x * blockDim.x + threadIdx.x;
  const int kvTot = B_ * KC_ * C_;
  if (idx < kvTot) {
    const int b = idx / (KC_ * C_);
    const int r = idx % (KC_ * C_);
    float s = 0.0f;
    for (int c = 0; c < NCHUNK; ++c)
      s += pkv[(long)(b * NCHUNK + c) * (KC_ * C_) + r];
    kvf[idx] = s;
  } else if (idx < kvTot + B_ * KC_) {
    const int i2 = idx - kvTot;
    const int b = i2 / KC_;
    const int r = i2 % KC_;
    float s = 0.0f;
    for (int c = 0; c < NCHUNK; ++c) s += pz[(b * NCHUNK + c) * KC_ + r];
    zf[i2] = s;
  }
}

// ---------------------------------------------------------------------------
// Kernel 3: qm = fm(Wq@x+bq); out = gamma * (kv^T @ qm) / max(qm.z,1e-6) + x
// ---------------------------------------------------------------------------
__global__ __launch_bounds__(256) void k3_out(
    const float* __restrict__ x, const float* __restrict__ Wq,
    const float* __restrict__ bq, const float* __restrict__ kvf,
    const float* __restrict__ zf, const float* __restrict__ gamma,
    float* __restrict__ out) {
  __shared__ float xs[C_ * XS_LD];
  __shared__ float qs[KC_ * KS_LD];
  __shared__ float invs[TPX];

  const int tid = threadIdx.x;
  const int lane = tid & 31;
  const int wave = tid >> 5;
  const int chunk = blockIdx.x;
  const int b = blockIdx.y;
  const int n0 = chunk * CHUNKPX;

  const int am = lane & 15;
  const int kg = (lane >> 4) * 2;
  const int nn = lane & 15;
  const int mh = (lane >> 4) * 8;

  const float g = gamma[0];
  const float* xb = x + (long)b * C_ * N_;
  const float* kvb = kvf + b * KC_ * C_;
  const float* zb = zf + b * KC_;

  for (int t = 0; t < TITER; ++t) {
    __syncthreads();
    stage_x_tile(xb, n0 + t * TPX, xs, tid);
    __syncthreads();
    // qm = fm(Wq @ xs + bq): 4 tiles, waves 0-3
    if (wave < 4) {
      const int mb = 16 * (wave & 1);
      const int nb = 16 * (wave >> 1);
      v8f acc = {};
      for (int kb = 0; kb < C_; kb += 4) {
        const int k = kb + kg;
        v2f a, bf;
        a.x = Wq[(mb + am) * C_ + k];
        a.y = Wq[(mb + am) * C_ + k + 1];
        bf.x = xs[k * XS_LD + nb + nn];
        bf.y = xs[(k + 1) * XS_LD + nb + nn];
        acc = wmma4(a, bf, acc);
      }
      for (int r = 0; r < 8; ++r) {
        const int m = mb + mh + r;
        qs[m * KS_LD + nb + nn] = fm_elu1(acc[r] + bq[m]);
      }
    }
    __syncthreads();
    // per-pixel normalizer
    if (tid < TPX) {
      float s = 0.0f;
      for (int kc = 0; kc < KC_; ++kc) s += qs[kc * KS_LD + tid] * zb[kc];
      invs[tid] = 1.0f / fmaxf(s, 1e-6f);
    }
    __syncthreads();
    // out tile [256][32] = kv^T @ qm : 32 tiles, 4 per wave
    for (int j = 0; j < 4; ++j) {
      const int tt = wave * 4 + j;
      const int mb = 16 * (tt >> 1);
      const int nb = 16 * (tt & 1);
      v8f acc = {};
      for (int kb = 0; kb < KC_; kb += 4) {
        const int k = kb + kg;
        v2f a, bf;
        a.x = kvb[k * C_ + mb + am];        // A[m=vrow][k=kc] = kv[kc][vrow]
        a.y = kvb[(k + 1) * C_ + mb + am];
        bf.x = qs[k * KS_LD + nb + nn];
        bf.y = qs[(k + 1) * KS_LD + nb + nn];
        acc = wmma4(a, bf, acc);
      }
      const float iv = invs[nb + nn];
      for (int r = 0; r < 8; ++r) {
        const int m = mb + mh + r;
        const float val = g * (acc[r] * iv) + xs[m * XS_LD + nb + nn];
        out[((long)b * C_ + m) * N_ + n0 + t * TPX + nb + nn] = val;
      }
    }
  }
}

// ---------------------------------------------------------------------------
extern "C" void kernel_launch(void* const* d_in, const int* in_sizes, int n_in,
                              void* d_out, int out_size, void* d_ws,
                              size_t ws_size, hipStream_t stream) {
  const float* x = (const float*)d_in[0];
  const float* Wq = (const float*)d_in[1];
  const float* bq = (const float*)d_in[2];
  const float* Wk = (const float*)d_in[3];
  const float* bk = (const float*)d_in[4];
  const float* Wv = (const float*)d_in[5];
  const float* bv = (const float*)d_in[6];
  const float* gamma = (const float*)d_in[7];
  float* out = (float*)d_out;

  float* ws = (float*)d_ws;
  float* pkv = ws;                                        // [B][NCHUNK][32][256]
  float* pz = pkv + (size_t)B_ * NCHUNK * KC_ * C_;       // [B][NCHUNK][32]
  float* kvf = pz + (size_t)B_ * NCHUNK * KC_;            // [B][32][256]
  float* zf = kvf + (size_t)B_ * KC_ * C_;                // [B][32]

  dim3 grid(NCHUNK, B_);
  k1_kv<<<grid, 256, 0, stream>>>(x, Wk, bk, Wv, bv, pkv, pz);

  const int tot2 = B_ * KC_ * C_ + B_ * KC_;
  k2_reduce<<<(tot2 + 255) / 256, 256, 0, stream>>>(pkv, pz, kvf, zf);

  k3_out<<<grid, 256, 0, stream>>>(x, Wq, bq, kvf, zf, gamma, out);
}